// WarpingLayer_47236050321515
// MI455X (gfx1250) — compile-verified
//
#include <hip/hip_runtime.h>

// WarpingLayer for MI455X (gfx1250, wave32).
// Memory-bound scatter-warp: no matrix contraction -> no WMMA (arithmetic
// intensity ~0.1 flop/byte; HBM roofline ~252 MB / 23.3 TB/s ~= 10.8us).
// Strategy: 128-bit vmem everywhere, non-temporal hints for write-once /
// read-once planes so the 192MB L2 holds the re-read planes, and a
// deterministic two-pass scatter using L2 atomics (atomicMax of source index
// == last-write-wins in (i,j) order, channel-consistent winner). The error
// plane (out ch11) doubles as the collision-key buffer so no workspace is
// needed; k3 overwrites it with the real error at the end.
//
// Grid layout: batch index on blockIdx.y (SGPR, no integer divide), quad
// index on blockIdx.x; i/j derived by shift/mask since W=512 is a power of 2.

typedef float v4f __attribute__((ext_vector_type(4)));
typedef int   v4i __attribute__((ext_vector_type(4)));

namespace {
constexpr int BN    = 16;
constexpr int CIN   = 8;
constexpr int COUT  = 12;
constexpr int HH    = 384;
constexpr int WW    = 512;            // power of two -> j = p & 511, i = p >> 9
constexpr int PLANE = HH * WW;        // 196608 floats per channel plane
constexpr int PQ    = PLANE / 4;      // 49152 float4-quads per plane
constexpr long long IN_B  = (long long)CIN  * PLANE;   // batch stride in
constexpr long long OUT_B = (long long)COUT * PLANE;   // batch stride out
constexpr int TPB   = 256;            // 8 wave32 per block
constexpr int GX    = PQ / TPB;       // 192 blocks in x (exact)
}

// ---------------------------------------------------------------------------
// k1: copy im1/im2/flow into output, zero the warped planes, seed key plane
// ---------------------------------------------------------------------------
__global__ __launch_bounds__(TPB) void wl_k1_copy_zero(
    const float* __restrict__ x, float* __restrict__ out) {
  const int n = blockIdx.y;
  const int q = blockIdx.x * TPB + threadIdx.x;     // [0, PQ)

  const v4f* in = reinterpret_cast<const v4f*>(x + n * IN_B);
  v4f*       o  = reinterpret_cast<v4f*>(out + n * OUT_B);

  v4f a0 = in[0 * PQ + q];   // im1
  v4f a1 = in[1 * PQ + q];
  v4f a2 = in[2 * PQ + q];
  v4f b0 = in[3 * PQ + q];   // im2
  v4f b1 = in[4 * PQ + q];
  v4f b2 = in[5 * PQ + q];
  v4f f0 = in[6 * PQ + q];   // flow
  v4f f1 = in[7 * PQ + q];
  v4f z  = (v4f){0.f, 0.f, 0.f, 0.f};

  // ch0-2 (im1 copy) and ch9-10 (flow copy): never re-read -> NT stores
  __builtin_nontemporal_store(a0, &o[0 * PQ + q]);
  __builtin_nontemporal_store(a1, &o[1 * PQ + q]);
  __builtin_nontemporal_store(a2, &o[2 * PQ + q]);
  __builtin_nontemporal_store(f0, &o[9 * PQ + q]);
  __builtin_nontemporal_store(f1, &o[10 * PQ + q]);

  // ch3-5 (warped: re-written by k2b, re-read by k3) and ch6-8 (im2:
  // re-read by k3): regular temporal stores so they stay resident in L2.
  o[3 * PQ + q] = z;
  o[4 * PQ + q] = z;
  o[5 * PQ + q] = z;
  o[6 * PQ + q] = b0;
  o[7 * PQ + q] = b1;
  o[8 * PQ + q] = b2;

  // ch11 used as int key plane for scatter resolution: init to -1
  v4i neg1 = (v4i){-1, -1, -1, -1};
  v4i* key = reinterpret_cast<v4i*>(out + n * OUT_B + 11LL * PLANE);
  key[q] = neg1;
}

// ---------------------------------------------------------------------------
// k2a: scatter key pass -> atomicMax(source linear pixel index) at target
// ---------------------------------------------------------------------------
__global__ __launch_bounds__(TPB) void wl_k2a_scatter_key(
    const float* __restrict__ x, float* __restrict__ out) {
  const int n = blockIdx.y;
  const int q = blockIdx.x * TPB + threadIdx.x;
  const int p = q << 2;               // pixel offset within plane
  const int i = p >> 9;               // row   (W = 512)
  const int j = p & (WW - 1);         // col of first pixel (row-contiguous)

  const float* xb  = x + n * IN_B;
  int*         key = reinterpret_cast<int*>(out + n * OUT_B + 11LL * PLANE);

  v4f fx = *reinterpret_cast<const v4f*>(xb + 6LL * PLANE + p);  // flow ch0: cols
  v4f fy = *reinterpret_cast<const v4f*>(xb + 7LL * PLANE + p);  // flow ch1: rows
  const float fi = (float)i;

#pragma unroll
  for (int k = 0; k < 4; ++k) {
    // v_rndne_f32 == round-half-to-even == jnp.round
    float tyf = __builtin_rintf(fx[k] + (float)(j + k));
    float txf = __builtin_rintf(fy[k] + fi);
    // float-domain bounds check: NaN/inf/negative/too-large all drop
    if (txf >= 0.f && txf < (float)HH && tyf >= 0.f && tyf < (float)WW) {
      int t = ((int)txf << 9) | (int)tyf;
      atomicMax(&key[t], p + k);      // last-write-wins in (i,j) order
    }
  }
}

// ---------------------------------------------------------------------------
// k2b: scatter write pass -> unique winner writes its 3 im1 channels
// ---------------------------------------------------------------------------
__global__ __launch_bounds__(TPB) void wl_k2b_scatter_write(
    const float* __restrict__ x, float* __restrict__ out) {
  const int n = blockIdx.y;
  const int q = blockIdx.x * TPB + threadIdx.x;
  const int p = q << 2;
  const int i = p >> 9;
  const int j = p & (WW - 1);

  const float* xb  = x + n * IN_B;
  float*       wr  = out + n * OUT_B + 3LL * PLANE;              // warped base
  const int*   key = reinterpret_cast<const int*>(out + n * OUT_B + 11LL * PLANE);

  v4f fx = *reinterpret_cast<const v4f*>(xb + 6LL * PLANE + p);
  v4f fy = *reinterpret_cast<const v4f*>(xb + 7LL * PLANE + p);
  v4f c0 = *reinterpret_cast<const v4f*>(xb + 0LL * PLANE + p);  // im1
  v4f c1 = *reinterpret_cast<const v4f*>(xb + 1LL * PLANE + p);
  v4f c2 = *reinterpret_cast<const v4f*>(xb + 2LL * PLANE + p);
  const float fi = (float)i;

#pragma unroll
  for (int k = 0; k < 4; ++k) {
    float tyf = __builtin_rintf(fx[k] + (float)(j + k));
    float txf = __builtin_rintf(fy[k] + fi);
    if (txf >= 0.f && txf < (float)HH && tyf >= 0.f && tyf < (float)WW) {
      int t = ((int)txf << 9) | (int)tyf;
      if (key[t] == p + k) {          // this source won the collision
        wr[0 * PLANE + t] = c0[k];
        wr[1 * PLANE + t] = c1[k];
        wr[2 * PLANE + t] = c2[k];
      }
    }
  }
}

// ---------------------------------------------------------------------------
// k3: error = sqrt(sum_c (warped - im2)^2); overwrites the key plane (ch11)
// ---------------------------------------------------------------------------
__global__ __launch_bounds__(TPB) void wl_k3_error(float* __restrict__ out) {
  const int n = blockIdx.y;
  const int q = blockIdx.x * TPB + threadIdx.x;

  const v4f* o = reinterpret_cast<const v4f*>(out + n * OUT_B);
  // last use of these planes -> non-temporal loads (free L2 early)
  v4f w0 = __builtin_nontemporal_load(&o[3 * PQ + q]);
  v4f w1 = __builtin_nontemporal_load(&o[4 * PQ + q]);
  v4f w2 = __builtin_nontemporal_load(&o[5 * PQ + q]);
  v4f m0 = __builtin_nontemporal_load(&o[6 * PQ + q]);
  v4f m1 = __builtin_nontemporal_load(&o[7 * PQ + q]);
  v4f m2 = __builtin_nontemporal_load(&o[8 * PQ + q]);

  v4f e;
#pragma unroll
  for (int k = 0; k < 4; ++k) {
    float d0 = w0[k] - m0[k];
    float d1 = w1[k] - m1[k];
    float d2 = w2[k] - m2[k];
    e[k] = __builtin_sqrtf(d0 * d0 + d1 * d1 + d2 * d2);
  }

  v4f* ov = reinterpret_cast<v4f*>(out + n * OUT_B);
  __builtin_nontemporal_store(e, &ov[11 * PQ + q]);   // write-once plane
}

// ---------------------------------------------------------------------------
extern "C" void kernel_launch(void* const* d_in, const int* in_sizes, int n_in,
                              void* d_out, int out_size, void* d_ws, size_t ws_size,
                              hipStream_t stream) {
  (void)in_sizes; (void)n_in; (void)out_size; (void)d_ws; (void)ws_size;
  const float* x   = reinterpret_cast<const float*>(d_in[0]);
  float*       out = reinterpret_cast<float*>(d_out);

  dim3 blk(TPB);
  dim3 grd(GX, BN);                    // 192 x 16 blocks, exact cover

  wl_k1_copy_zero     <<<grd, blk, 0, stream>>>(x, out);
  wl_k2a_scatter_key  <<<grd, blk, 0, stream>>>(x, out);
  wl_k2b_scatter_write<<<grd, blk, 0, stream>>>(x, out);
  wl_k3_error         <<<grd, blk, 0, stream>>>(out);
}